// ModelNew_72902774882653
// MI455X (gfx1250) — compile-verified
//
#include <hip/hip_runtime.h>
#include <stdint.h>

#define THREADS 256
#define NBIN    4096
#define CHUNK   (NBIN / THREADS)   // 16 bins per thread in the scan phase
#define GCAP    2048               // max gathered boundary-bin elements
#define TILE    1024               // floats per async tile: 256 lanes * 16B
#define EPS_TOPP 1e-8f

typedef float vfloat4 __attribute__((ext_vector_type(4)));  // clang vector: OK for nontemporal builtins

// order-preserving float -> uint key (ascending); compare descending on key
__device__ __forceinline__ uint32_t f2key(float f) {
  uint32_t u = __float_as_uint(f);
  return u ^ ((u >> 31) ? 0xFFFFFFFFu : 0x80000000u);
}
__device__ __forceinline__ float key2f(uint32_t k) {
  uint32_t u = (k & 0x80000000u) ? (k ^ 0x80000000u) : ~k;
  return __uint_as_float(u);
}

__global__ __launch_bounds__(THREADS)
void topkp_sample_kernel(const float* __restrict__ logits,
                         const int*   __restrict__ top_ks,
                         const float* __restrict__ top_ps,
                         const float* __restrict__ q,
                         long long*   __restrict__ out_idx,
                         float*       __restrict__ out_masked,
                         int V)
{
  __shared__ __align__(16) float stage[2][TILE];   // async staging, 2 x 4KB
  __shared__ uint32_t h_cnt[NBIN];
  __shared__ float    h_sum[NBIN];
  __shared__ uint32_t gkey[GCAP];
  __shared__ int      gcnt;
  __shared__ uint32_t partC[THREADS];
  __shared__ float    partS[THREADS];
  __shared__ float    red_val[THREADS];
  __shared__ int      red_idx[THREADS];
  __shared__ int      s_binK, s_binP, s_c, s_npad, s_r;
  __shared__ uint32_t s_cntAbove, s_tkKey, s_thresh;
  __shared__ float    s_sumAboveK, s_sumAboveP, s_T;

  const int tid = threadIdx.x;
  const int row = blockIdx.x;
  const float NEG_INF = __uint_as_float(0xFF800000u);

  const float* rowL = logits + (size_t)row * V;
  const float* rowQ = q      + (size_t)row * V;
  float*       rowO = out_masked + (size_t)row * V;
  const int n4 = V >> 2;
  const vfloat4* L4 = (const vfloat4*)rowL;

  // ------- gather + exact bitonic sort of one boundary bin (L2-hot scan) ---
  auto gather_sort = [&](int bin) {
    if (tid == 0) gcnt = 0;
    __syncthreads();
    for (int i = tid; i < n4; i += THREADS) {            // B128 loads, hits L2
      vfloat4 v = L4[i];
      #pragma unroll
      for (int e = 0; e < 4; ++e) {
        uint32_t ky = f2key(v[e]);
        if ((int)(ky >> 20) == bin) {
          int p = atomicAdd(&gcnt, 1);
          if (p < GCAP) gkey[p] = ky;
        }
      }
    }
    for (int i = (n4 << 2) + tid; i < V; i += THREADS) { // tail (none for V=128000)
      uint32_t ky = f2key(rowL[i]);
      if ((int)(ky >> 20) == bin) {
        int p = atomicAdd(&gcnt, 1);
        if (p < GCAP) gkey[p] = ky;
      }
    }
    __syncthreads();
    if (tid == 0) {
      int c = gcnt < GCAP ? gcnt : GCAP;
      if (c < 1) c = 1;
      int np = 1; while (np < c) np <<= 1;
      s_c = c; s_npad = np;
    }
    __syncthreads();
    const int c = s_c, np = s_npad;
    for (int i = c + tid; i < np; i += THREADS) gkey[i] = 0u;  // pad with -inf key
    __syncthreads();
    for (int kk = 2; kk <= np; kk <<= 1) {
      for (int j = kk >> 1; j > 0; j >>= 1) {
        for (int i = tid; i < np; i += THREADS) {
          int ixj = i ^ j;
          if (ixj > i) {
            uint32_t a = gkey[i], b = gkey[ixj];
            bool descBlk = ((i & kk) == 0);
            if (descBlk ? (a < b) : (a > b)) { gkey[i] = b; gkey[ixj] = a; }
          }
        }
        __syncthreads();
      }
    }
  };

  // ------- Pass 1: 12-bit histogram {count, sum(exp)}; async double-buffer -
  for (int i = tid; i < NBIN; i += THREADS) { h_cnt[i] = 0u; h_sum[i] = 0.0f; }
  __syncthreads();

  const int nTiles = V / TILE;                      // 128000/1024 = 125
  const uint32_t lds0 = (uint32_t)(uintptr_t)&stage[0][tid * 4];
  const uint32_t lds1 = (uint32_t)(uintptr_t)&stage[1][tid * 4];

  if (nTiles > 0) {
    uint64_t ga0 = (uint64_t)(uintptr_t)(rowL + tid * 4);
    asm volatile("global_load_async_to_lds_b128 %0, %1, off"
                 :: "v"(lds0), "v"(ga0) : "memory");
  }
  for (int t = 0; t < nTiles; ++t) {
    if (t + 1 < nTiles) {
      uint32_t dst = ((t + 1) & 1) ? lds1 : lds0;
      uint64_t ga = (uint64_t)(uintptr_t)(rowL + (size_t)(t + 1) * TILE + tid * 4);
      asm volatile("global_load_async_to_lds_b128 %0, %1, off"
                   :: "v"(dst), "v"(ga) : "memory");
      asm volatile("s_wait_asynccnt 0x1" ::: "memory");   // tile t complete
    } else {
      asm volatile("s_wait_asynccnt 0x0" ::: "memory");
    }
    const vfloat4 v = *(const vfloat4*)&stage[t & 1][tid * 4];
    #pragma unroll
    for (int e = 0; e < 4; ++e) {
      uint32_t b = f2key(v[e]) >> 20;
      atomicAdd(&h_cnt[b], 1u);            // ds_add_u32
      atomicAdd(&h_sum[b], __expf(v[e]));  // ds_add_f32
    }
  }
  for (int i = nTiles * TILE + tid; i < V; i += THREADS) {   // tail
    float f = rowL[i];
    uint32_t b = f2key(f) >> 20;
    atomicAdd(&h_cnt[b], 1u);
    atomicAdd(&h_sum[b], __expf(f));
  }
  __syncthreads();

  // parallel chunk partials over descending bin positions
  {
    uint32_t pc = 0; float ps = 0.0f;
    #pragma unroll
    for (int i = 0; i < CHUNK; ++i) {
      int b = (NBIN - 1) - (tid * CHUNK + i);
      pc += h_cnt[b]; ps += h_sum[b];
    }
    partC[tid] = pc; partS[tid] = ps;
  }
  __syncthreads();

  const int K = top_ks[row];

  // ------- top-k boundary bin (exact integer crossing) ---------------------
  if (tid == 0) {
    uint32_t cum = 0; float sum = 0.0f; int chunk = THREADS - 1;
    for (int c = 0; c < THREADS; ++c) {
      if (cum + partC[c] >= (uint32_t)K) { chunk = c; break; }
      cum += partC[c]; sum += partS[c];
    }
    int bsel = (NBIN - 1) - (chunk * CHUNK + CHUNK - 1);
    for (int i = 0; i < CHUNK; ++i) {
      int b = (NBIN - 1) - (chunk * CHUNK + i);
      uint32_t cc = h_cnt[b];
      if (cum + cc >= (uint32_t)K) { bsel = b; break; }
      cum += cc; sum += h_sum[b];
    }
    s_binK = bsel; s_cntAbove = cum; s_sumAboveK = sum;
  }
  __syncthreads();

  gather_sort(s_binK);

  if (tid == 0) {
    int r = K - (int)s_cntAbove;          // rank of t_k inside this bin, 1-based
    if (r < 1) r = 1;
    if (r > s_c) r = s_c;
    s_tkKey = gkey[r - 1];
    s_r = r;
  }
  __syncthreads();

  // Z = exp-mass of the exact top-k set; T = (p+eps)*Z
  {
    float zp = 0.0f;
    for (int i = tid; i < s_r; i += THREADS) zp += __expf(key2f(gkey[i]));
    red_val[tid] = zp;
  }
  __syncthreads();
  if (tid == 0) {
    float z = s_sumAboveK;
    for (int i = 0; i < THREADS; ++i) z += red_val[i];
    s_T = (top_ps[row] + EPS_TOPP) * z;
  }
  __syncthreads();

  // ------- top-p boundary bin (by exp-mass) --------------------------------
  if (tid == 0) {
    const float T = s_T;
    float cum = 0.0f; int chunk = -1;
    for (int c = 0; c < THREADS; ++c) {
      if (cum + partS[c] > T) { chunk = c; break; }
      cum += partS[c];
    }
    if (chunk < 0) { s_binP = -1; s_sumAboveP = cum; }   // everything p-kept
    else {
      int bsel = (NBIN - 1) - (chunk * CHUNK + CHUNK - 1);
      for (int i = 0; i < CHUNK; ++i) {
        int b = (NBIN - 1) - (chunk * CHUNK + i);
        float ss = h_sum[b];
        if (cum + ss > T) { bsel = b; break; }
        cum += ss;
      }
      s_binP = bsel; s_sumAboveP = cum;
    }
  }
  __syncthreads();

  const uint32_t tkKey = s_tkKey;
  if (s_binP >= 0 && s_binP != s_binK) gather_sort(s_binP);  // else reuse sorted bin
  __syncthreads();

  if (tid == 0) {
    uint32_t tp = 0u;
    if (s_binP >= 0) {
      float cum = s_sumAboveP;
      const float T = s_T;
      const int c = s_c;
      for (int i = 0; i < c; ++i) {
        uint32_t ky = gkey[i];
        if (ky < tkKey) break;             // top-k already dominates below this
        if (cum <= T) tp = ky; else break; // keep while mass-before <= T
        cum += __expf(key2f(ky));
      }
    }
    s_thresh = (tkKey > tp) ? tkKey : tp;
  }
  __syncthreads();

  // ------- final pass: mask + Gumbel argmax --------------------------------
  // logits re-read: L2-resident (131MB < 192MB L2). q read + masked store are
  // streamed non-temporally so they don't evict the logits working set.
  const uint32_t th = s_thresh;
  float best = NEG_INF; int bidx = 0;
  const vfloat4* Q4 = (const vfloat4*)rowQ;
  vfloat4*       O4 = (vfloat4*)rowO;
  for (int i = tid; i < n4; i += THREADS) {
    vfloat4 lv = L4[i];
    vfloat4 qv = __builtin_nontemporal_load(&Q4[i]);
    vfloat4 mv;
    #pragma unroll
    for (int e = 0; e < 4; ++e) {
      float l = lv[e];
      bool keep = f2key(l) >= th;
      mv[e] = keep ? l : NEG_INF;
      if (keep) {
        float sc = l - __logf(qv[e]);      // argmax probs/q == argmax l - log q
        int gi = i * 4 + e;
        if (sc > best) { best = sc; bidx = gi; }
      }
    }
    __builtin_nontemporal_store(mv, &O4[i]);
  }
  for (int i = (n4 << 2) + tid; i < V; i += THREADS) {  // tail (none here)
    float l = rowL[i];
    bool keep = f2key(l) >= th;
    rowO[i] = keep ? l : NEG_INF;
    if (keep) {
      float sc = l - __logf(rowQ[i]);
      if (sc > best) { best = sc; bidx = i; }
    }
  }
  red_val[tid] = best; red_idx[tid] = bidx;
  __syncthreads();
  if (tid == 0) {
    float b = red_val[0]; int bi = red_idx[0];
    for (int i = 1; i < THREADS; ++i) {
      float v = red_val[i];
      if (v > b || (v == b && red_idx[i] < bi)) { b = v; bi = red_idx[i]; }
    }
    out_idx[row] = (long long)bi;
  }
}

extern "C" void kernel_launch(void* const* d_in, const int* in_sizes, int n_in,
                              void* d_out, int out_size, void* d_ws, size_t ws_size,
                              hipStream_t stream) {
  (void)n_in; (void)out_size; (void)d_ws; (void)ws_size;
  const float* logits = (const float*)d_in[0];
  const int*   top_ks = (const int*)d_in[1];
  const float* top_ps = (const float*)d_in[2];
  const float* q      = (const float*)d_in[3];
  const int N = in_sizes[1];                 // 256 rows
  const int V = in_sizes[0] / N;             // 128000 vocab
  long long* out_idx    = (long long*)d_out;              // int64[N] first
  float*     out_masked = (float*)d_out + 2 * (size_t)N;  // then f32[N*V]
  topkp_sample_kernel<<<N, THREADS, 0, stream>>>(logits, top_ks, top_ps, q,
                                                 out_idx, out_masked, V);
}